// InternalOptimizer_14035953124009
// MI455X (gfx1250) — compile-verified
//
#include <hip/hip_runtime.h>

typedef __attribute__((ext_vector_type(2))) float v2f;
typedef __attribute__((ext_vector_type(4))) float v4f;
typedef __attribute__((ext_vector_type(8))) float v8f;

namespace {
constexpr int   kD    = 1024;  // matrix dimension
constexpr int   kB    = 64;    // batch
constexpr int   kTile = 16;    // WMMA M tile
constexpr float kEta  = 0.01f;
}

__device__ __forceinline__ float readlane_f32(float v, int lane) {
  return __builtin_bit_cast(
      float, __builtin_amdgcn_readlane(__builtin_bit_cast(int, v), lane));
}

// One wave (32 lanes) per 16-row tile of one batch's W.
// Phase 1: Wx for 16 rows via V_WMMA_F32_16X16X4_F32 (B broadcasts x across N).
// Phase 2: out = W - (Wx + eta*gL) x^T, coalesced float4 streams.
//   - W re-read is NT (last use: let L2 drop it after),
//   - out store is NT (write-once, never read: don't pollute the 192MB L2),
//   so L2 capacity stays dedicated to the phase1->phase2 W reuse window.
__global__ __launch_bounds__(32) void InternalOptimizer_rank1_wmma(
    const float* __restrict__ W,
    const float* __restrict__ x,
    const float* __restrict__ gL,
    float* __restrict__ out)
{
  const int tile = blockIdx.x;             // 0 .. B*(D/16)-1
  const int b    = tile >> 6;              // 64 tiles per batch
  const int r0   = (tile & 63) * kTile;    // first row of this tile
  const int lane = threadIdx.x;            // wave32 lane

  const float* Wb  = W   + (size_t)b * kD * kD;
  const float* xb  = x   + (size_t)b * kD;
  const float* gLb = gL  + (size_t)b * kD;
  float*       Ob  = out + (size_t)b * kD * kD;

  // ---- Phase 1: acc[m][n] += W[m][k] * x[k] via 16x16x4 f32 WMMA ----
  // A layout (ISA 7.12.2, 32-bit A 16x4): lanes 0-15 -> K=0 (v0), K=1 (v1);
  // lanes 16-31 -> K=2 (v0), K=3 (v1); M = lane & 15.
  const int  m    = lane & 15;
  const bool hi   = lane >= 16;
  const int  kofs = hi ? 2 : 0;
  const float* rowA = Wb + (size_t)(r0 + m) * kD;

  v8f acc = {};
  #pragma unroll 4
  for (int k0 = 0; k0 < kD; k0 += 4) {
    v2f a;
    a.x = rowA[k0 + kofs];
    a.y = rowA[k0 + kofs + 1];
    // uniform x chunk -> scalar loads; B[k][n] = x[k] for all n
    float x0 = xb[k0 + 0], x1 = xb[k0 + 1], x2 = xb[k0 + 2], x3 = xb[k0 + 3];
    v2f bm;
    bm.x = hi ? x2 : x0;
    bm.y = hi ? x3 : x1;
    acc = __builtin_amdgcn_wmma_f32_16x16x4_f32(
        /*neg_a=*/false, a, /*neg_b=*/false, bm,
        /*c_mod=*/(short)0, acc, /*reuse_a=*/false, /*reuse_b=*/false);
  }

  // C/D layout: VGPR r holds M=r (lanes 0-15) and M=8+r (lanes 16-31);
  // every N column equals the dot product -> v_readlane from lanes 0 / 16
  // gives SGPR-resident per-row scales (no LDS traffic).
  float s[16];
  #pragma unroll
  for (int r = 0; r < 8; ++r) {
    s[r]     = readlane_f32(acc[r], 0)  + kEta * gLb[r0 + r];
    s[r + 8] = readlane_f32(acc[r], 16) + kEta * gLb[r0 + r + 8];
  }

  // ---- Phase 2: out[m][j] = W[m][j] - s[m] * x[j], float4 coalesced ----
  #pragma unroll 2
  for (int cc = 0; cc < kD; cc += 128) {        // 32 lanes * 4 floats
    const int col = cc + lane * 4;
    const v4f xv = *reinterpret_cast<const v4f*>(xb + col);
    #pragma unroll
    for (int r = 0; r < 16; ++r) {
      const size_t ofs = (size_t)(r0 + r) * kD + col;
      const v4f wv =
          __builtin_nontemporal_load(reinterpret_cast<const v4f*>(Wb + ofs));
      v4f ov;
      ov.x = fmaf(-s[r], xv.x, wv.x);
      ov.y = fmaf(-s[r], xv.y, wv.y);
      ov.z = fmaf(-s[r], xv.z, wv.z);
      ov.w = fmaf(-s[r], xv.w, wv.w);
      __builtin_nontemporal_store(ov, reinterpret_cast<v4f*>(Ob + ofs));
    }
  }
}

extern "C" void kernel_launch(void* const* d_in, const int* in_sizes, int n_in,
                              void* d_out, int out_size, void* d_ws, size_t ws_size,
                              hipStream_t stream) {
  (void)in_sizes; (void)n_in; (void)out_size; (void)d_ws; (void)ws_size;
  const float* W  = (const float*)d_in[0];
  const float* x  = (const float*)d_in[1];
  const float* gL = (const float*)d_in[2];
  float* out = (float*)d_out;

  const int tiles = kB * (kD / kTile);  // 4096 waves, one per 16-row tile
  InternalOptimizer_rank1_wmma<<<dim3(tiles), dim3(32), 0, stream>>>(W, x, gL, out);
}